// Model_73461120631380
// MI455X (gfx1250) — compile-verified
//
#include <hip/hip_runtime.h>
#include <math.h>

#define NQ   20000
#define NS   8192
#define C    112
#define C2   224
#define CLS  20
#define TILES (NQ / 16)          // 1250, exact
#define WAVES_PER_BLOCK 4

// per-wave LDS region (floats)
#define WREG    3712
#define SSTRIDE 20               // score tile row stride (bank-conflict-free, 16B aligned rows)
#define XSTRIDE 226              // X[16 x 224] row stride (even -> b64 loads, spreads banks)
#define HSTRIDE 114              // H / Y2 row stride
#define H_OFF   0
#define Y2_OFF  1856

typedef float v2f __attribute__((ext_vector_type(2)));
typedef float v8f __attribute__((ext_vector_type(8)));

// ---------------- Kernel A1: logits[j] = feat_s7[j,:] . w_attn ----------------
__global__ void attn_logits(const float* __restrict__ feat_s7,
                            const float* __restrict__ w_attn,
                            float* __restrict__ logits) {
  int j = blockIdx.x * blockDim.x + threadIdx.x;
  if (j >= NS) return;
  const float* row = feat_s7 + (long)j * C;
  float acc = 0.f;
#pragma unroll 8
  for (int c = 0; c < C; ++c) acc += row[c] * w_attn[c];
  logits[j] = acc;
}

// ---------- Kernel A2: softmax pool -> g[C] -> gate[c] = sigmoid(g[c]) ----------
__global__ void attn_pool_gate(const float* __restrict__ feat_s7,
                               const float* __restrict__ logits,
                               float* __restrict__ ebuf,
                               float* __restrict__ gate) {
  __shared__ float red[256];
  __shared__ float bcast[2];
  const int t = threadIdx.x;

  float m = -1e30f;
  for (int j = t; j < NS; j += 256) m = fmaxf(m, logits[j]);
  red[t] = m; __syncthreads();
  for (int s = 128; s > 0; s >>= 1) { if (t < s) red[t] = fmaxf(red[t], red[t + s]); __syncthreads(); }
  if (t == 0) bcast[0] = red[0];
  __syncthreads();
  m = bcast[0];

  float sum = 0.f;
  for (int j = t; j < NS; j += 256) { float e = __expf(logits[j] - m); ebuf[j] = e; sum += e; }
  __syncthreads();
  red[t] = sum; __syncthreads();
  for (int s = 128; s > 0; s >>= 1) { if (t < s) red[t] += red[t + s]; __syncthreads(); }
  if (t == 0) bcast[1] = red[0];
  __syncthreads();
  const float inv = 1.0f / bcast[1];

  if (t < C) {
    float acc = 0.f;
    for (int j = 0; j < NS; ++j) acc += ebuf[j] * feat_s7[(long)j * C + t];
    float g = acc * inv;
    gate[t] = 1.0f / (1.0f + __expf(-g));
  }
}

// branchless ascending top-3 insert (v0 <= v1 <= v2v)
#define TOP3_INS(val, idxv) {                                         \
    float s_ = (val); int id_ = (idxv);                               \
    bool c0_ = s_ < v0, c1_ = s_ < v1, c2_ = s_ < v2v;                \
    v2v = c1_ ? v1 : (c2_ ? s_  : v2v); i2 = c1_ ? i1 : (c2_ ? id_ : i2); \
    v1  = c0_ ? v0 : (c1_ ? s_  : v1 ); i1 = c0_ ? i0 : (c1_ ? id_ : i1); \
    v0  = c0_ ? s_ : v0;                i0 = c0_ ? id_ : i0;          }

// ---------------- Fused main kernel: one wave per 16-query tile ----------------
__global__ __launch_bounds__(128) void fused_main(
    const float* __restrict__ coords_q, const float* __restrict__ coords_s,
    const float* __restrict__ x7,       const float* __restrict__ feat_s,
    const float* __restrict__ gate,     const float* __restrict__ W1,
    const float* __restrict__ gammaW,   const float* __restrict__ betaW,
    const float* __restrict__ W2,       const float* __restrict__ W_cls,
    const float* __restrict__ b_cls,    float* __restrict__ out)
{
  __shared__ float lds[WAVES_PER_BLOCK * WREG];
  const int lane = threadIdx.x & 31;
  const int wave = threadIdx.x >> 5;
  const int tile = blockIdx.x * WAVES_PER_BLOCK + wave;
  if (tile >= TILES) return;                 // wave-uniform

  float* WL = lds + wave * WREG;
  const int r15  = lane & 15;
  const int half = lane >> 4;
  const int qbase = tile * 16;

  // ---- Phase 1: score = s^2 - 2 q.s via V_WMMA_F32_16X16X4_F32, running top-3 ----
  const float qx = coords_q[(qbase + r15) * 3 + 0];
  const float qy = coords_q[(qbase + r15) * 3 + 1];
  const float qz = coords_q[(qbase + r15) * 3 + 2];
  v2f a;   // A[16x4]: row M=r15, K = 2*half + {0,1};  A = [-2qx,-2qy,-2qz, 1]
  a.x = half ? (-2.f * qz) : (-2.f * qx);
  a.y = half ? 1.0f        : (-2.f * qy);

  float v0 = 1e30f, v1 = 1e30f, v2v = 1e30f;
  int   i0 = 0, i1 = 0, i2 = 0;

  for (int n0 = 0; n0 < NS; n0 += 16) {
    __builtin_prefetch(coords_s + (n0 + 128) * 3, 0, 1);
    const int j = n0 + r15;
    const float sx = coords_s[j * 3 + 0];
    const float sy = coords_s[j * 3 + 1];
    const float sz = coords_s[j * 3 + 2];
    const float s2 = sx * sx + sy * sy + sz * sz;
    v2f b;   // B[4x16]: col N=r15, K = 2*half + {0,1};  B col = [sx,sy,sz,s2]^T
    b.x = half ? sz : sx;
    b.y = half ? s2 : sy;
    v8f c = {0.f, 0.f, 0.f, 0.f, 0.f, 0.f, 0.f, 0.f};
    c = __builtin_amdgcn_wmma_f32_16x16x4_f32(false, a, false, b, (short)0, c, false, false);

    // scatter C tile (VGPR r -> M = r + 8*half, N = r15) to LDS
#pragma unroll
    for (int r = 0; r < 8; ++r)
      WL[(r + 8 * half) * SSTRIDE + r15] = c[r];
    asm volatile("" ::: "memory");           // keep DS order; LDS is in-order per wave

    // lane owns query row r15, scans cols [8*half, 8*half+8)
    const float4 sA = *(const float4*)(WL + r15 * SSTRIDE + 8 * half);
    const float4 sB = *(const float4*)(WL + r15 * SSTRIDE + 8 * half + 4);

    // wave-uniform prune: skip the serial insert chain when no lane can improve
    const float m8 = fminf(fminf(fminf(sA.x, sA.y), fminf(sA.z, sA.w)),
                           fminf(fminf(sB.x, sB.y), fminf(sB.z, sB.w)));
    if (__any(m8 < v2v)) {
      const int base = n0 + 8 * half;
      TOP3_INS(sA.x, base + 0); TOP3_INS(sA.y, base + 1);
      TOP3_INS(sA.z, base + 2); TOP3_INS(sA.w, base + 3);
      TOP3_INS(sB.x, base + 4); TOP3_INS(sB.y, base + 5);
      TOP3_INS(sB.z, base + 6); TOP3_INS(sB.w, base + 7);
    }
    asm volatile("" ::: "memory");
  }

  // merge the two half-lists for row r15 (symmetric: both halves end with full top-3)
  {
    float pv0 = __shfl(v0, lane + 16), pv1 = __shfl(v1, lane + 16), pv2 = __shfl(v2v, lane + 16);
    int   pi0 = __shfl(i0, lane + 16), pi1 = __shfl(i1, lane + 16), pi2 = __shfl(i2,  lane + 16);
    TOP3_INS(pv0, pi0); TOP3_INS(pv1, pi1); TOP3_INS(pv2, pi2);
  }
  // softmax(-score) over top-3 (shift-invariant: dropping q^2 is exact)
  const float e1 = __expf(v0 - v1), e2 = __expf(v0 - v2v);
  const float inv = 1.0f / (1.0f + e1 + e2);
  const float w0 = inv, w1 = e1 * inv, w2 = e2 * inv;

  // ---- Phase 2: build X[16 x 224] = [ x7*gate | matched ] in LDS ----
  for (int r = 0; r < 16; ++r) {
    const float rw0 = __shfl(w0, r), rw1 = __shfl(w1, r), rw2 = __shfl(w2, r);
    const int   ri0 = __shfl(i0, r), ri1 = __shfl(i1, r), ri2 = __shfl(i2, r);
    const int qi = qbase + r;
    for (int t = lane; t < C2; t += 32) {
      float v;
      if (t < C) {
        v = x7[(long)qi * C + t] * gate[t];
      } else {
        const int cc = t - C;
        v = rw0 * feat_s[(long)ri0 * C + cc]
          + rw1 * feat_s[(long)ri1 * C + cc]
          + rw2 * feat_s[(long)ri2 * C + cc];
      }
      WL[r * XSTRIDE + t] = v;
    }
  }
  asm volatile("" ::: "memory");

  // ---- Phase 3: GEMM1  Y[16x112] = X[16x224] @ W1[224x112] ----
  v8f acc[7];
#pragma unroll
  for (int n = 0; n < 7; ++n) acc[n] = (v8f){0.f, 0.f, 0.f, 0.f, 0.f, 0.f, 0.f, 0.f};
  for (int k0 = 0; k0 < C2; k0 += 4) {
    const int kk = k0 + 2 * half;
    const v2f a1 = *(const v2f*)(WL + r15 * XSTRIDE + kk);
#pragma unroll
    for (int n = 0; n < 7; ++n) {
      v2f bv;
      bv.x = W1[(long)kk * C + n * 16 + r15];
      bv.y = W1[(long)(kk + 1) * C + n * 16 + r15];
      acc[n] = __builtin_amdgcn_wmma_f32_16x16x4_f32(false, a1, false, bv, (short)0, acc[n], false, false);
    }
  }

  // ---- Phase 4: H = relu(Y*gamma + beta), stage to LDS (A-layout source) ----
#pragma unroll
  for (int n = 0; n < 7; ++n) {
    const int ch = n * 16 + r15;
    const float gm = gammaW[ch], bt = betaW[ch];
#pragma unroll
    for (int r = 0; r < 8; ++r) {
      const float h = fmaxf(acc[n][r] * gm + bt, 0.f);
      WL[H_OFF + (r + 8 * half) * HSTRIDE + ch] = h;
    }
  }
  asm volatile("" ::: "memory");

  // ---- Phase 5: GEMM2  Y2[16x112] = H @ W2[112x112] ----
  v8f acc2[7];
#pragma unroll
  for (int n = 0; n < 7; ++n) acc2[n] = (v8f){0.f, 0.f, 0.f, 0.f, 0.f, 0.f, 0.f, 0.f};
  for (int k0 = 0; k0 < C; k0 += 4) {
    const int kk = k0 + 2 * half;
    const v2f a2 = *(const v2f*)(WL + H_OFF + r15 * HSTRIDE + kk);
#pragma unroll
    for (int n = 0; n < 7; ++n) {
      v2f bv;
      bv.x = W2[(long)kk * C + n * 16 + r15];
      bv.y = W2[(long)(kk + 1) * C + n * 16 + r15];
      acc2[n] = __builtin_amdgcn_wmma_f32_16x16x4_f32(false, a2, false, bv, (short)0, acc2[n], false, false);
    }
  }
#pragma unroll
  for (int n = 0; n < 7; ++n) {
    const int ch = n * 16 + r15;
#pragma unroll
    for (int r = 0; r < 8; ++r)
      WL[Y2_OFF + (r + 8 * half) * HSTRIDE + ch] = acc2[n][r];
  }
  asm volatile("" ::: "memory");

  // ---- Phase 6: GEMM3  out[16x20] = Y2 @ W_cls[112x20] + b_cls ----
  v8f acc3[2];
  acc3[0] = (v8f){0.f, 0.f, 0.f, 0.f, 0.f, 0.f, 0.f, 0.f};
  acc3[1] = acc3[0];
  for (int k0 = 0; k0 < C; k0 += 4) {
    const int kk = k0 + 2 * half;
    const v2f a3 = *(const v2f*)(WL + Y2_OFF + r15 * HSTRIDE + kk);
#pragma unroll
    for (int n = 0; n < 2; ++n) {
      const int col = n * 16 + r15;
      v2f bv;
      bv.x = (col < CLS) ? W_cls[(long)kk * CLS + col] : 0.f;
      bv.y = (col < CLS) ? W_cls[(long)(kk + 1) * CLS + col] : 0.f;
      acc3[n] = __builtin_amdgcn_wmma_f32_16x16x4_f32(false, a3, false, bv, (short)0, acc3[n], false, false);
    }
  }
#pragma unroll
  for (int n = 0; n < 2; ++n) {
    const int col = n * 16 + r15;
    if (col < CLS) {
      const float bb = b_cls[col];
#pragma unroll
      for (int r = 0; r < 8; ++r)
        out[(long)(qbase + r + 8 * half) * CLS + col] = acc3[n][r] + bb;
    }
  }
}

extern "C" void kernel_launch(void* const* d_in, const int* in_sizes, int n_in,
                              void* d_out, int out_size, void* d_ws, size_t ws_size,
                              hipStream_t stream) {
  const float* coords_q = (const float*)d_in[0];
  const float* coords_s = (const float*)d_in[1];
  const float* x7       = (const float*)d_in[2];
  const float* feat_s7  = (const float*)d_in[3];
  const float* feat_s   = (const float*)d_in[4];
  const float* w_attn   = (const float*)d_in[5];
  const float* W1       = (const float*)d_in[6];
  const float* gammaW   = (const float*)d_in[7];
  const float* betaW    = (const float*)d_in[8];
  const float* W2       = (const float*)d_in[9];
  const float* W_cls    = (const float*)d_in[10];
  const float* b_cls    = (const float*)d_in[11];
  float* out = (float*)d_out;

  float* ws     = (float*)d_ws;
  float* logits = ws;            // NS floats
  float* ebuf   = ws + NS;       // NS floats
  float* gate   = ws + 2 * NS;   // C floats

  attn_logits<<<NS / 256, 256, 0, stream>>>(feat_s7, w_attn, logits);
  attn_pool_gate<<<1, 256, 0, stream>>>(feat_s7, logits, ebuf, gate);

  const int blocks = (TILES + WAVES_PER_BLOCK - 1) / WAVES_PER_BLOCK;
  fused_main<<<blocks, 128, 0, stream>>>(coords_q, coords_s, x7, feat_s, gate,
                                         W1, gammaW, betaW, W2, W_cls, b_cls, out);
}